// LightGCNLayer_30554397343957
// MI455X (gfx1250) — compile-verified
//
#include <hip/hip_runtime.h>
#include <stdint.h>

#define N_FEAT 128

// ---- gfx1250 async-to-LDS availability probe ------------------------------
#if __has_builtin(__builtin_amdgcn_global_load_async_to_lds_b64) && \
    __has_builtin(__builtin_amdgcn_global_load_async_to_lds_b32)
#define HAVE_ASYNC_LDS 1
#else
#define HAVE_ASYNC_LDS 0
#endif

#if __has_builtin(__builtin_amdgcn_s_wait_asynccnt)
#define WAIT_ASYNC(n) __builtin_amdgcn_s_wait_asynccnt(n)
#else
#define WAIT_ASYNC(n) asm volatile("s_wait_asynccnt %0" ::"i"(n) : "memory")
#endif

// Exact pointer types the async builtins expect (clang prints AS1 as
// "__device__", AS3 as "__shared__"): b64 wants int2-vector pointees,
// b32 wants int pointees; both non-const.
typedef int v2i __attribute__((vector_size(8)));
typedef __attribute__((address_space(1))) v2i* gptr_v2i;
typedef __attribute__((address_space(3))) v2i* lptr_v2i;
typedef __attribute__((address_space(1))) int* gptr_i32;
typedef __attribute__((address_space(3))) int* lptr_i32;

// ---------------------------------------------------------------------------
// Zero-fill d_out (harness poisons it with 0xAA; segment-sum needs zeros).
// ---------------------------------------------------------------------------
__global__ __launch_bounds__(256) void lightgcn_zero_kernel(float4* __restrict__ out,
                                                            int n4) {
  int i = blockIdx.x * 256 + threadIdx.x;
  if (i < n4) out[i] = make_float4(0.f, 0.f, 0.f, 0.f);
}

// ---------------------------------------------------------------------------
// Stage one 32-edge tile's metadata into a per-wave LDS buffer.
// Async path: 3 x GLOBAL_LOAD_ASYNC_TO_LDS (b64 src, b64 dst, b32 w), ASYNCcnt.
// Fallback: NT vector loads + LDS stores.
// ---------------------------------------------------------------------------
__device__ __forceinline__ void stage_tile(const long long* __restrict__ esrc,
                                           const long long* __restrict__ edst,
                                           const float* __restrict__ ew,
                                           int base, int n_edges, int lane,
                                           long long* lsrc, long long* ldst,
                                           float* lw) {
  int idx = base + lane;
  if (idx >= n_edges) idx = n_edges - 1;   // clamp: keeps EXEC full, dup rows
                                           // never consumed past tile_edges
#if HAVE_ASYNC_LDS
  __builtin_amdgcn_global_load_async_to_lds_b64(
      (gptr_v2i)(uintptr_t)(esrc + idx),
      (lptr_v2i)(uintptr_t)(lsrc + lane), 0, 0);
  __builtin_amdgcn_global_load_async_to_lds_b64(
      (gptr_v2i)(uintptr_t)(edst + idx),
      (lptr_v2i)(uintptr_t)(ldst + lane), 0, 0);
  __builtin_amdgcn_global_load_async_to_lds_b32(
      (gptr_i32)(uintptr_t)(ew + idx),
      (lptr_i32)(uintptr_t)(lw + lane), 0, 0);
#else
  lsrc[lane] = __builtin_nontemporal_load(&esrc[idx]);
  ldst[lane] = __builtin_nontemporal_load(&edst[idx]);
  lw[lane]   = __builtin_nontemporal_load(&ew[idx]);
#endif
}

// ---------------------------------------------------------------------------
// COO scatter: one wave32 per 32-edge tile, grid-strided, with double-buffered
// async LDS staging of the edge stream. Per tile: readlane-broadcast src/dst/w
// to SGPRs, 4x global_prefetch_b8 per lane covers all 32 gather rows, then a
// coalesced float4 row gather + 4 no-return global_atomic_add_f32 per lane.
// ---------------------------------------------------------------------------
__global__ __launch_bounds__(256) void lightgcn_scatter_kernel(
    const float* __restrict__ features,
    const float* __restrict__ edge_w,
    const long long* __restrict__ edge_src,
    const long long* __restrict__ edge_dst,
    float* __restrict__ out,
    int n_edges) {
  // Per-wave double-buffered metadata tiles: 2 bufs x 8 waves x 32 edges.
  __shared__ long long lds_src[2][8][32];
  __shared__ long long lds_dst[2][8][32];
  __shared__ float     lds_w[2][8][32];

  const int lane   = threadIdx.x & 31;
  const int wv     = threadIdx.x >> 5;                 // wave in block (0..7)
  const int wave   = (blockIdx.x << 3) | wv;           // global wave id
  const int nwaves = gridDim.x << 3;
  const int ntiles = (n_edges + 31) >> 5;
  if (wave >= ntiles) return;

  // Prologue: stage first tile into buffer 0.
  stage_tile(edge_src, edge_dst, edge_w, wave << 5, n_edges, lane,
             lds_src[0][wv], lds_dst[0][wv], lds_w[0][wv]);

  int buf = 0;
  for (int t = wave; t < ntiles; t += nwaves, buf ^= 1) {
    const int base = t << 5;
    const int nxt  = t + nwaves;

    // Stage next tile into the other buffer, then wait for CURRENT tile:
    // async loads complete in order, so asynccnt<=3 => current's 3 are done.
    if (nxt < ntiles) {
      stage_tile(edge_src, edge_dst, edge_w, nxt << 5, n_edges, lane,
                 lds_src[buf ^ 1][wv], lds_dst[buf ^ 1][wv], lds_w[buf ^ 1][wv]);
#if HAVE_ASYNC_LDS
      WAIT_ASYNC(3);
#endif
    } else {
#if HAVE_ASYNC_LDS
      WAIT_ASYNC(0);
#endif
    }

    // Pull this lane's edge metadata out of LDS (ds_load), truncate i64->i32.
    const int   my_src = (int)lds_src[buf][wv][lane];   // < 50000, fits i32
    const int   my_dst = (int)lds_dst[buf][wv][lane];
    const float my_w   = lds_w[buf][wv][lane];

    // Prefetch all 32 gather rows of this tile (512 B row = 4 cachelines).
    const char* frow = (const char*)features + (size_t)my_src * (N_FEAT * 4);
    __builtin_prefetch(frow +   0, 0, 3);
    __builtin_prefetch(frow + 128, 0, 3);
    __builtin_prefetch(frow + 256, 0, 3);
    __builtin_prefetch(frow + 384, 0, 3);

    const int tile_edges = min(32, n_edges - base);     // wave-uniform
    for (int e = 0; e < tile_edges; ++e) {
      // Broadcast edge e's metadata to SGPRs (wave32 readlane, e uniform).
      const int   s_src = __builtin_amdgcn_readlane(my_src, e);
      const int   s_dst = __builtin_amdgcn_readlane(my_dst, e);
      const float w     = __uint_as_float(
          (unsigned)__builtin_amdgcn_readlane((int)__float_as_uint(my_w), e));

      // Coalesced 512B row gather: 32 lanes x float4 (global_load_b128).
      const float4 f =
          *(const float4*)(features + (size_t)s_src * N_FEAT + (lane << 2));

      float* o = out + (size_t)s_dst * N_FEAT + (lane << 2);
      atomicAdd(o + 0, w * f.x);   // no-return global_atomic_add_f32
      atomicAdd(o + 1, w * f.y);
      atomicAdd(o + 2, w * f.z);
      atomicAdd(o + 3, w * f.w);
    }
  }
}

// ---------------------------------------------------------------------------
// Launch
// ---------------------------------------------------------------------------
extern "C" void kernel_launch(void* const* d_in, const int* in_sizes, int n_in,
                              void* d_out, int out_size, void* d_ws, size_t ws_size,
                              hipStream_t stream) {
  const float*     features = (const float*)d_in[0];
  const float*     edge_w   = (const float*)d_in[1];
  const long long* edge_src = (const long long*)d_in[2];
  const long long* edge_dst = (const long long*)d_in[3];
  float*           out      = (float*)d_out;

  const int n_edges = in_sizes[1];
  const int n_out4  = out_size >> 2;   // 50000*128/4 float4 elements

  lightgcn_zero_kernel<<<(n_out4 + 255) / 256, 256, 0, stream>>>(
      (float4*)out, n_out4);

  // ~4 tiles per wave so the double-buffer pipeline has depth to overlap.
  const int ntiles       = (n_edges + 31) / 32;          // 25000
  const int waves_needed = (ntiles + 3) / 4;             // 6250
  const int blocks       = (waves_needed + 7) / 8;       // 782 blocks x 8 waves
  lightgcn_scatter_kernel<<<blocks, 256, 0, stream>>>(
      features, edge_w, edge_src, edge_dst, out, n_edges);
}